// SE3Transformer_45148696216855
// MI455X (gfx1250) — compile-verified
//
#include <hip/hip_runtime.h>
#include <math.h>

// ---------------- problem constants ----------------
#define N_NODES 20000
#define N_EDGES 320000
#define C0I 32
#define C1I 16
#define C0O 64
#define C1O 16
// tiles: N/16 = 1250 exactly, E/16 = 20000 exactly

typedef float v2f  __attribute__((ext_vector_type(2)));
typedef float v4f  __attribute__((ext_vector_type(4)));
typedef float v8f  __attribute__((ext_vector_type(8)));

// ---------------- workspace layout (floats) ----------------
// q0[N,64] | logits/e[E] | m_enc[N] (uint) | z[N] | packed weights
constexpr int OFF_Q0   = 0;
constexpr int OFF_E    = OFF_Q0 + N_NODES * 64;   // 1,280,000
constexpr int OFF_M    = OFF_E + N_EDGES;         // 1,600,000
constexpr int OFF_Z    = OFF_M + N_NODES;         // 1,620,000
constexpr int OFF_PWQ0 = OFF_Z + N_NODES;         // 1,640,000  (32x64)
constexpr int OFF_PWS0 = OFF_PWQ0 + 2048;         // Wself0 (32x64)
constexpr int OFF_PWK  = OFF_PWS0 + 2048;         // [Wk00;Wk10] (48x64)
constexpr int OFF_PWV  = OFF_PWK + 3072;          // [Wv00;Wv10] (48x64)
constexpr int OFF_PW01 = OFF_PWV + 3072;          // Wv01 (32x16)
constexpr int OFF_PW11 = OFF_PW01 + 512;          // Wv11 (16x16)
constexpr int OFF_PWS1 = OFF_PW11 + 256;          // Wself1 (16x16)
// total floats = OFF_PWS1 + 256 = 1,651,264  (~6.6 MB)

// ---------------- fp32 WMMA 16x16x4 ----------------
__device__ __forceinline__ v8f wmma4(v2f a, v2f b, v8f c) {
  // D = A(16x4 f32) * B(4x16 f32) + C(16x16 f32), exact fp32
  return __builtin_amdgcn_wmma_f32_16x16x4_f32(false, a, false, b, (short)0, c,
                                               false, false);
}

// monotonic float<->uint encoding for atomicMax-based segment max
__device__ __forceinline__ unsigned encodeF(float f) {
  unsigned u = __float_as_uint(f);
  return (u & 0x80000000u) ? ~u : (u | 0x80000000u);
}
__device__ __forceinline__ float decodeF(unsigned u) {
  return __uint_as_float((u & 0x80000000u) ? (u & 0x7fffffffu) : ~u);
}

// pack W[K][Nc] so that B-fragment pairs (k, k+1) per column are adjacent:
// dst[(k>>1)*2*Nc + n*2 + (k&1)] = W[k][n]   (k offset by krow0 for concat)
__device__ __forceinline__ void packB(const float* __restrict__ W,
                                      float* __restrict__ dst, int K, int Nc,
                                      int krow0, int tid, int nth) {
  for (int idx = tid; idx < K * Nc; idx += nth) {
    int k = idx / Nc + krow0;
    int n = idx % Nc;
    dst[(k >> 1) * (2 * Nc) + n * 2 + (k & 1)] = W[idx];
  }
}

// ================= kernel 0: pack weights + init reductions =================
__global__ void se3_prep_kernel(const float* Wq0, const float* Wk00,
                                const float* Wk10, const float* Wv00,
                                const float* Wv10, const float* Wv01,
                                const float* Wv11, const float* Wself0,
                                const float* Wself1, float* ws) {
  int tid = blockIdx.x * blockDim.x + threadIdx.x;
  int nth = gridDim.x * blockDim.x;
  unsigned* m_enc = (unsigned*)(ws + OFF_M);
  float* z = ws + OFF_Z;
  for (int i = tid; i < N_NODES; i += nth) {
    m_enc[i] = 0u;   // encodes the minimum
    z[i] = 0.0f;
  }
  packB(Wq0,    ws + OFF_PWQ0, 32, 64, 0,  tid, nth);
  packB(Wself0, ws + OFF_PWS0, 32, 64, 0,  tid, nth);
  packB(Wk00,   ws + OFF_PWK,  32, 64, 0,  tid, nth);
  packB(Wk10,   ws + OFF_PWK,  16, 64, 32, tid, nth);
  packB(Wv00,   ws + OFF_PWV,  32, 64, 0,  tid, nth);
  packB(Wv10,   ws + OFF_PWV,  16, 64, 32, tid, nth);
  packB(Wv01,   ws + OFF_PW01, 32, 16, 0,  tid, nth);
  packB(Wv11,   ws + OFF_PW11, 16, 16, 0,  tid, nth);
  packB(Wself1, ws + OFF_PWS1, 16, 16, 0,  tid, nth);
}

// ================= kernel 1: node-level GEMMs (q0, self terms) ==============
// one wave (32 threads) per 16-node tile; 1250 blocks
__global__ __launch_bounds__(32) void se3_node_kernel(
    const float* __restrict__ x0, const float* __restrict__ x1,
    float* __restrict__ ws, float* __restrict__ out) {
  const int l    = threadIdx.x & 31;
  const int col  = l & 15;       // A-row index == D-col index == lane&15
  const int half = l >> 4;
  const int tile = blockIdx.x;
  const int rowA = tile * 16 + col;

  const float* pWq  = ws + OFF_PWQ0;
  const float* pWs0 = ws + OFF_PWS0;
  const float* pWs1 = ws + OFF_PWS1;
  float* q0   = ws + OFF_Q0;
  float* out0 = out;
  float* out1 = out + N_NODES * 64;

  // q0 = x0 @ Wq0 ; out0 = x0 @ Wself0 (K=32, 4 column tiles)
  for (int which = 0; which < 2; ++which) {
    const float* pB = which ? pWs0 : pWq;
    float* dbuf     = which ? out0 : q0;
    for (int n0 = 0; n0 < 64; n0 += 16) {
      v8f acc = {};
#pragma unroll
      for (int kb = 0; kb < 32; kb += 4) {
        int keff = kb + half * 2;
        v2f a = *(const v2f*)(x0 + rowA * 32 + keff);
        v2f b = *(const v2f*)(pB + (keff >> 1) * 128 + (n0 + col) * 2);
        acc = wmma4(a, b, acc);
      }
#pragma unroll
      for (int v = 0; v < 8; ++v) {
        int node = tile * 16 + v + 8 * half;
        dbuf[node * 64 + n0 + col] = acc[v];
      }
    }
  }

  // out1 = x1 @ Wself1 per xyz component (K=16, one column tile)
  for (int d = 0; d < 3; ++d) {
    v8f acc = {};
#pragma unroll
    for (int kb = 0; kb < 16; kb += 4) {
      int keff = kb + half * 2;
      v2f a;
      a.x = x1[rowA * 48 + keff * 3 + d];
      a.y = x1[rowA * 48 + (keff + 1) * 3 + d];
      v2f b = *(const v2f*)(pWs1 + (keff >> 1) * 32 + col * 2);
      acc = wmma4(a, b, acc);
    }
#pragma unroll
    for (int v = 0; v < 8; ++v) {
      int node = tile * 16 + v + 8 * half;
      out1[node * 48 + col * 3 + d] = acc[v];
    }
  }
}

// ================= kernel 2: edge pass A — k0 GEMM + logits + seg-max =======
// 8 waves/block, one 16-edge tile per wave; 2500 blocks (exact)
__global__ __launch_bounds__(256) void se3_edgeA_kernel(
    const float* __restrict__ x0, const float* __restrict__ x1,
    const float* __restrict__ pos, const int* __restrict__ ei,
    float* __restrict__ ws) {
  __shared__ float featA[8][16][48];   // [edge][x0s(32) | x1dot(16)]
  __shared__ float Y1s[8][16][3];
  __shared__ int   sidx[8][2][16];     // [src|dst][edge]

  const int w    = threadIdx.x >> 5;
  const int l    = threadIdx.x & 31;
  const int e    = l & 15;
  const int half = l >> 4;
  const int tile = blockIdx.x * 8 + w;
  const int e0   = tile * 16;

  const float* q0     = ws + OFF_Q0;
  float* logits       = ws + OFF_E;
  unsigned* m_enc     = (unsigned*)(ws + OFF_M);
  const float* pWK    = ws + OFF_PWK;

  if (half == 0) {
    int s = ei[e0 + e];
    int d = ei[N_EDGES + e0 + e];
    sidx[w][0][e] = s;
    sidx[w][1][e] = d;
    float rx = pos[d * 3 + 0] - pos[s * 3 + 0];
    float ry = pos[d * 3 + 1] - pos[s * 3 + 1];
    float rz = pos[d * 3 + 2] - pos[s * 3 + 2];
    float inv = 1.0f / (sqrtf(rx * rx + ry * ry + rz * rz) + 1e-8f);
    Y1s[w][e][0] = rx * inv;
    Y1s[w][e][1] = ry * inv;
    Y1s[w][e][2] = rz * inv;
  }
  __syncthreads();

  const int s = sidx[w][0][e];
  // stage x0s (each half covers 16 of 32 channels, float4 loads)
#pragma unroll
  for (int i = 0; i < 4; ++i) {
    v4f v = *(const v4f*)(x0 + s * 32 + half * 16 + i * 4);
    *(v4f*)&featA[w][e][half * 16 + i * 4] = v;
  }
  // x1dot = x1s . Y1 (each half covers 8 of 16 channels)
  {
    float y0 = Y1s[w][e][0], y1 = Y1s[w][e][1], y2 = Y1s[w][e][2];
    float buf[24];
#pragma unroll
    for (int i = 0; i < 6; ++i) {
      v4f v = *(const v4f*)(x1 + s * 48 + half * 24 + i * 4);
      buf[i * 4 + 0] = v.x; buf[i * 4 + 1] = v.y;
      buf[i * 4 + 2] = v.z; buf[i * 4 + 3] = v.w;
    }
#pragma unroll
    for (int cc = 0; cc < 8; ++cc)
      featA[w][e][32 + half * 8 + cc] =
          buf[cc * 3 + 0] * y0 + buf[cc * 3 + 1] * y1 + buf[cc * 3 + 2] * y2;
  }
  __syncthreads();

  int de[8];
#pragma unroll
  for (int v = 0; v < 8; ++v) de[v] = sidx[w][1][v + 8 * half];

  float part[8] = {0, 0, 0, 0, 0, 0, 0, 0};
  for (int n0 = 0; n0 < 64; n0 += 16) {
    v8f acc = {};
#pragma unroll
    for (int kb = 0; kb < 48; kb += 4) {
      int keff = kb + half * 2;
      v2f a = *(const v2f*)&featA[w][e][keff];
      v2f b = *(const v2f*)(pWK + (keff >> 1) * 128 + (n0 + e) * 2);
      acc = wmma4(a, b, acc);
    }
#pragma unroll
    for (int v = 0; v < 8; ++v)
      part[v] += acc[v] * q0[de[v] * 64 + n0 + e];
  }
  // reduce over the 16 lanes of each half (each lane held one column group)
#pragma unroll
  for (int off = 1; off < 16; off <<= 1)
#pragma unroll
    for (int v = 0; v < 8; ++v)
      part[v] += __shfl_xor(part[v], off, 16);

  if (e < 8) {
    int erow = e + 8 * half;
    float lg = part[e] * 0.125f;  // 1/sqrt(64)
    logits[e0 + erow] = lg;
    atomicMax(&m_enc[sidx[w][1][erow]], encodeF(lg));
  }
}

// ================= kernel 3: edge pass B — exp + segment sum ================
__global__ void se3_edgeB_kernel(const int* __restrict__ ei,
                                 float* __restrict__ ws) {
  int i = blockIdx.x * blockDim.x + threadIdx.x;
  if (i >= N_EDGES) return;
  float* logits   = ws + OFF_E;
  unsigned* m_enc = (unsigned*)(ws + OFF_M);
  float* z        = ws + OFF_Z;
  int d = ei[N_EDGES + i];
  float ev = expf(logits[i] - decodeF(m_enc[d]));
  logits[i] = ev;  // buffer now holds e
  atomicAdd(&z[d], ev);
}

// ================= kernel 4: edge pass C — values + scatter =================
__global__ __launch_bounds__(256) void se3_edgeC_kernel(
    const float* __restrict__ x0, const float* __restrict__ x1,
    const float* __restrict__ pos, const int* __restrict__ ei,
    float* __restrict__ ws, float* __restrict__ out) {
  __shared__ float featA[8][16][48];
  __shared__ float x1sT[8][3][16][16];  // [d][edge][channel]
  __shared__ float Y1s[8][16][3];
  __shared__ float alf[8][16];
  __shared__ int   sidx[8][2][16];

  const int w    = threadIdx.x >> 5;
  const int l    = threadIdx.x & 31;
  const int e    = l & 15;
  const int half = l >> 4;
  const int tile = blockIdx.x * 8 + w;
  const int e0   = tile * 16;

  const float* earr = ws + OFF_E;
  const float* z    = ws + OFF_Z;
  const float* pWV  = ws + OFF_PWV;
  const float* pW01 = ws + OFF_PW01;
  const float* pW11 = ws + OFF_PW11;
  float* out0 = out;
  float* out1 = out + N_NODES * 64;

  if (half == 0) {
    int s = ei[e0 + e];
    int d = ei[N_EDGES + e0 + e];
    sidx[w][0][e] = s;
    sidx[w][1][e] = d;
    float rx = pos[d * 3 + 0] - pos[s * 3 + 0];
    float ry = pos[d * 3 + 1] - pos[s * 3 + 1];
    float rz = pos[d * 3 + 2] - pos[s * 3 + 2];
    float inv = 1.0f / (sqrtf(rx * rx + ry * ry + rz * rz) + 1e-8f);
    Y1s[w][e][0] = rx * inv;
    Y1s[w][e][1] = ry * inv;
    Y1s[w][e][2] = rz * inv;
    alf[w][e] = earr[e0 + e] / (z[d] + 1e-8f);
  }
  __syncthreads();

  const int s = sidx[w][0][e];
#pragma unroll
  for (int i = 0; i < 4; ++i) {
    v4f v = *(const v4f*)(x0 + s * 32 + half * 16 + i * 4);
    *(v4f*)&featA[w][e][half * 16 + i * 4] = v;
  }
  {
    float y0 = Y1s[w][e][0], y1 = Y1s[w][e][1], y2 = Y1s[w][e][2];
    float buf[24];
#pragma unroll
    for (int i = 0; i < 6; ++i) {
      v4f v = *(const v4f*)(x1 + s * 48 + half * 24 + i * 4);
      buf[i * 4 + 0] = v.x; buf[i * 4 + 1] = v.y;
      buf[i * 4 + 2] = v.z; buf[i * 4 + 3] = v.w;
    }
#pragma unroll
    for (int cc = 0; cc < 8; ++cc) {
      int c = half * 8 + cc;
      float a0 = buf[cc * 3 + 0], a1 = buf[cc * 3 + 1], a2 = buf[cc * 3 + 2];
      x1sT[w][0][e][c] = a0;
      x1sT[w][1][e][c] = a1;
      x1sT[w][2][e][c] = a2;
      featA[w][e][32 + c] = a0 * y0 + a1 * y1 + a2 * y2;
    }
  }
  __syncthreads();

  float alr[8];
  int de[8];
#pragma unroll
  for (int v = 0; v < 8; ++v) {
    int rv = v + 8 * half;
    alr[v] = alf[w][rv];
    de[v]  = sidx[w][1][rv];
  }

  // v0 = [x0s|x1dot] @ [Wv00;Wv10]  (K=48, 4 column tiles), scaled by alpha
  for (int n0 = 0; n0 < 64; n0 += 16) {
    v8f acc = {};
#pragma unroll
    for (int kb = 0; kb < 48; kb += 4) {
      int keff = kb + half * 2;
      v2f a = *(const v2f*)&featA[w][e][keff];
      v2f b = *(const v2f*)(pWV + (keff >> 1) * 128 + (n0 + e) * 2);
      acc = wmma4(a, b, acc);
    }
#pragma unroll
    for (int v = 0; v < 8; ++v)
      atomicAdd(&out0[de[v] * 64 + n0 + e], acc[v] * alr[v]);
  }

  // c01 = x0s @ Wv01 (K=32, 16 cols) — shared across xyz components
  v8f c01 = {};
#pragma unroll
  for (int kb = 0; kb < 32; kb += 4) {
    int keff = kb + half * 2;
    v2f a = *(const v2f*)&featA[w][e][keff];
    v2f b = *(const v2f*)(pW01 + (keff >> 1) * 32 + e * 2);
    c01 = wmma4(a, b, c01);
  }

  // v1[:,f,d] = x1s[:,:,d] @ Wv11 + c01 * Y1[:,d], scaled by alpha, scatter
  for (int d = 0; d < 3; ++d) {
    v8f cd = {};
#pragma unroll
    for (int kb = 0; kb < 16; kb += 4) {
      int keff = kb + half * 2;
      v2f a = *(const v2f*)&x1sT[w][d][e][keff];
      v2f b = *(const v2f*)(pW11 + (keff >> 1) * 32 + e * 2);
      cd = wmma4(a, b, cd);
    }
#pragma unroll
    for (int v = 0; v < 8; ++v) {
      int rv = v + 8 * half;
      float val = (cd[v] + c01[v] * Y1s[w][rv][d]) * alr[v];
      atomicAdd(&out1[de[v] * 48 + e * 3 + d], val);
    }
  }
}

// ============================ launch ============================
extern "C" void kernel_launch(void* const* d_in, const int* in_sizes, int n_in,
                              void* d_out, int out_size, void* d_ws,
                              size_t ws_size, hipStream_t stream) {
  const float* x0     = (const float*)d_in[0];
  const float* x1     = (const float*)d_in[1];
  const float* pos    = (const float*)d_in[2];
  const int*   ei     = (const int*)d_in[3];
  const float* Wq0    = (const float*)d_in[4];
  const float* Wk00   = (const float*)d_in[5];
  const float* Wk10   = (const float*)d_in[6];
  const float* Wv00   = (const float*)d_in[7];
  const float* Wv10   = (const float*)d_in[8];
  const float* Wv01   = (const float*)d_in[9];
  const float* Wv11   = (const float*)d_in[10];
  const float* Wself0 = (const float*)d_in[11];
  const float* Wself1 = (const float*)d_in[12];
  float* ws  = (float*)d_ws;
  float* out = (float*)d_out;

  se3_prep_kernel<<<96, 256, 0, stream>>>(Wq0, Wk00, Wk10, Wv00, Wv10, Wv01,
                                          Wv11, Wself0, Wself1, ws);
  se3_node_kernel<<<N_NODES / 16, 32, 0, stream>>>(x0, x1, ws, out);
  se3_edgeA_kernel<<<N_EDGES / (16 * 8), 256, 0, stream>>>(x0, x1, pos, ei, ws);
  se3_edgeB_kernel<<<(N_EDGES + 255) / 256, 256, 0, stream>>>(ei, ws);
  se3_edgeC_kernel<<<N_EDGES / (16 * 8), 256, 0, stream>>>(x0, x1, pos, ei, ws,
                                                           out);
}